// EfficientLearnableInterconnect_33913061769291
// MI455X (gfx1250) — compile-verified
//
#include <hip/hip_runtime.h>

// ---------------------------------------------------------------------------
// EfficientLearnableInterconnect: block-diagonal (2x2 blocks) GEMM, f32.
//   out[b, 2n+o] = sum_m x[b, 2n+m] * softmax_m(c[n, m, o])
// Memory-bound (0.5 FLOP/byte); WMMA f32 16x16x4 used on 16x16 block-diagonal
// weight tiles (8x redundant FLOPs, free under the HBM roofline).
// ---------------------------------------------------------------------------

typedef __attribute__((ext_vector_type(2))) float v2f;
typedef __attribute__((ext_vector_type(4))) float v4f;
typedef __attribute__((ext_vector_type(8))) float v8f;

#define BATCH     16384
#define LAYER     8192
#define NCOLT     (LAYER / 16)   // 512 column tiles
#define NROWT     (BATCH / 16)   // 1024 row tiles
#define LDS_STRIDE 20            // floats per row: 16B-aligned (80B) + conflict-free

__global__ __launch_bounds__(256) void
eli_wmma_kernel(const float* __restrict__ x,
                const float* __restrict__ c,
                float* __restrict__ out)
{
    // wave-private LDS tiles: 8 waves x 16 rows x 20 floats = 10240 B
    __shared__ float lds[8][16 * LDS_STRIDE];

    const int tid  = threadIdx.x;
    const int lane = tid & 31;
    const int wave = tid >> 5;

    const int col_tile  = blockIdx.x & (NCOLT - 1);  // 0..511
    const int row_group = blockIdx.x >> 9;           // 0..15
    const int col0      = col_tile << 4;

    // ---- build block-diagonal B operands once per wave --------------------
    // B (4x16, KxN) layout mirrors A/C/D: VGPR0 = rows {4k, 4k+2} in lane
    // halves, VGPR1 = rows {4k+1, 4k+3}.
    const int cN    = lane & 15;            // N (output column in tile)
    const int rhalf = (lane >> 4) << 1;     // 0 for lanes 0-15, 2 for 16-31
    const int nblk  = (col0 >> 1) + (cN >> 1);
    const int o     = cN & 1;

    // 2-element softmax over m: conn0 = sigmoid(e0 - e1), conn1 = 1 - conn0
    const float e0 = c[nblk * 4 + 0 + o];   // c[n][m=0][o]
    const float e1 = c[nblk * 4 + 2 + o];   // c[n][m=1][o]
    const float conn0 = 1.0f / (1.0f + __expf(e1 - e0));
    const float conn1 = 1.0f - conn0;

    v2f Bop[4];
#pragma unroll
    for (int kk = 0; kk < 4; ++kk) {
#pragma unroll
        for (int v = 0; v < 2; ++v) {
            const int r = 4 * kk + rhalf + v;             // K row of W
            float val = 0.0f;
            if ((r >> 1) == (cN >> 1))                    // on-diagonal block
                val = (r & 1) ? conn1 : conn0;            // m = r & 1
            Bop[kk][v] = val;
        }
    }

    // ---- per-wave addressing for loads / A reads / D stores ---------------
    float* myLds = &lds[wave][0];

    const int ldRow  = lane >> 2;          // 0..7  (rows loaded per lane pass)
    const int ldPart = (lane & 3) << 2;    // 0,4,8,12 (float4 within a row)

    const int aM    = lane & 15;           // A: M row for this lane
    const int khalf = (lane >> 4) << 1;    // A: K offset 0 or 2

    const int dRowOff = (lane >> 4) << 3;  // D: +8 rows for upper half-wave
    const int dCol    = col0 + (lane & 15);

#pragma unroll 1
    for (int i = 0; i < 8; ++i) {
        const int rt   = row_group * 64 + wave * 8 + i;   // row tile index
        const int row0 = rt << 4;

        // coalesced 16x16 f32 tile load: 2 x b128 per lane, nontemporal
        const float* src = x + (size_t)(row0 + ldRow) * LAYER + col0 + ldPart;
        const v4f t0 = __builtin_nontemporal_load((const v4f*)src);
        const v4f t1 = __builtin_nontemporal_load((const v4f*)(src + (size_t)8 * LAYER));

        *(v4f*)&myLds[ldRow * LDS_STRIDE + ldPart]       = t0;
        *(v4f*)&myLds[(ldRow + 8) * LDS_STRIDE + ldPart] = t1;

        // 16x16 x 16x16 block-diagonal product via 4 chained WMMA f32 16x16x4
        v8f acc = {};
#pragma unroll
        for (int kk = 0; kk < 4; ++kk) {
            v2f a;
            a[0] = myLds[aM * LDS_STRIDE + 4 * kk + khalf + 0];
            a[1] = myLds[aM * LDS_STRIDE + 4 * kk + khalf + 1];
            acc = __builtin_amdgcn_wmma_f32_16x16x4_f32(
                /*neg_a=*/false, a, /*neg_b=*/false, Bop[kk],
                /*c_mod=*/(short)0, acc, /*reuse_a=*/false, /*reuse_b=*/false);
        }

        // D layout: VGPR v -> out rows row0+v (lanes 0-15) / row0+8+v (16-31)
#pragma unroll
        for (int v = 0; v < 8; ++v) {
            float* dst = out + (size_t)(row0 + dRowOff + v) * LAYER + dCol;
            __builtin_nontemporal_store(acc[v], dst);
        }
    }
}

extern "C" void kernel_launch(void* const* d_in, const int* in_sizes, int n_in,
                              void* d_out, int out_size, void* d_ws, size_t ws_size,
                              hipStream_t stream)
{
    const float* x = (const float*)d_in[0];   // [16384, 8192]
    const float* c = (const float*)d_in[1];   // [4096, 2, 2]
    float* out     = (float*)d_out;           // [16384, 8192]

    dim3 grid(NCOLT * 16);   // 8192 blocks: 512 col tiles x 16 row groups
    dim3 block(256);         // 8 waves; each wave: 1 col tile x 8 row tiles
    hipLaunchKernelGGL(eli_wmma_kernel, grid, block, 0, stream, x, c, out);
}